// HOE_1_2_Module_27951647162476
// MI455X (gfx1250) — compile-verified
//
#include <hip/hip_runtime.h>
#include <hip/hip_bf16.h>

typedef __attribute__((ext_vector_type(2))) float v2f;
typedef __attribute__((ext_vector_type(8))) float v8f;

#define MF 128      // F = D = 128
#define MM 64       // M = 64
#define LDH 132     // padded LDS row stride (floats); 132%64=4 -> conflict-free A reads

// ---------------------------------------------------------------------------
// Kernel 1: per (b,n) slice, compute
//   mean[f]  = (1/64) sum_i x[i,f]
//   c[f]     = mean . W1[f,512:640] + b1[f]
//   meand[f] = mean . W1[f,384:512]
//   Dg[i,f]  = x_i . W1[f,  0:128] + meand[f]
//   U [i,f]  = x_i . W1[f,128:256]
//   V [j,f]  = x_j . W1[f,256:384]
// GEMMs done with v_wmma_f32_16x16x4_f32.
// ---------------------------------------------------------------------------
__global__ __launch_bounds__(256) void hoe_k1(
    const float* __restrict__ x, const float* __restrict__ W1,
    const float* __restrict__ b1,
    float* __restrict__ U, float* __restrict__ V,
    float* __restrict__ Dg, float* __restrict__ Cc)
{
    __shared__ float xs[MM][LDH];
    __shared__ float meanv[MF];
    __shared__ float meand[MF];

    const int bn = blockIdx.x;       // 0..31
    const int t  = threadIdx.x;      // 0..255
    const float* xg = x + (size_t)bn * MM * MF;

    // cooperative load x(64x128) into LDS (8 float4 per thread)
#pragma unroll
    for (int u = 0; u < 8; ++u) {
        int idx  = t + u * 256;          // float4 index 0..2047
        int row  = idx >> 5;             // 32 float4 per row
        int col4 = idx & 31;
        float4 vv = ((const float4*)xg)[idx];
        *(float4*)&xs[row][col4 * 4] = vv;
    }
    __syncthreads();

    if (t < MF) {
        float s = 0.f;
        for (int i2 = 0; i2 < MM; ++i2) s += xs[i2][t];
        meanv[t] = s * (1.f / 64.f);
    }
    __syncthreads();

    if (t < MF) {
        const float* w1r = W1 + t * (5 * MF);
        float cs = 0.f, ds = 0.f;
        for (int k = 0; k < MF; ++k) {
            float mk = meanv[k];
            cs += mk * w1r[512 + k];
            ds += mk * w1r[384 + k];
        }
        Cc[bn * MF + t] = cs + b1[t];
        meand[t] = ds;
    }
    __syncthreads();

    // WMMA GEMMs: out[i,f] = sum_k xs[i,k] * W1[f, off+k]
    const int lane = t & 31;
    const int wave = t >> 5;        // 0..7 -> col tile f0 = 16*wave
    const int n    = lane & 15;
    const int half = lane >> 4;
    const int f0   = wave * 16;

    for (int blk = 0; blk < 3; ++blk) {
        const int off = blk * MF;                 // 0 / 128 / 256
        // preload B fragments: B[k][n] = W1[f0+n, off+k]
        v2f breg[32];
        const float* wp = W1 + (size_t)(f0 + n) * (5 * MF) + off + 2 * half;
#pragma unroll
        for (int ks = 0; ks < 32; ++ks)
            breg[ks] = *(const v2f*)(wp + ks * 4);

        float* dst = (blk == 0 ? Dg : (blk == 1 ? U : V)) + (size_t)bn * MM * MF + f0 + n;
        const float addv = (blk == 0) ? meand[f0 + n] : 0.f;

#pragma unroll
        for (int it = 0; it < 4; ++it) {          // row tiles i0 = 16*it
            v8f acc = {};
#pragma unroll
            for (int ks = 0; ks < 32; ++ks) {
                // A[m][k] = xs[i0+m][4*ks + 2*half + {0,1}], m = lane%16
                v2f av = *(const v2f*)&xs[it * 16 + n][ks * 4 + 2 * half];
                acc = __builtin_amdgcn_wmma_f32_16x16x4_f32(
                        false, av, false, breg[ks], (short)0, acc, false, false);
            }
#pragma unroll
            for (int r = 0; r < 8; ++r) {
                int ii = it * 16 + r + 8 * half;  // C/D layout: row r + 8*half
                dst[ii * MF] = acc[r] + addv;
            }
        }
    }
}

// ---------------------------------------------------------------------------
// Kernel 2: one workgroup per (b,n,i).  Build h[j,f] = U[i]+V[j]+c+(j==i)Dg[i],
// LayerNorm over f, ReLU, +bias_p on diagonal, into LDS; then
// out[j,d] = h . W2[d,:] + b2[d] via v_wmma_f32_16x16x4_f32.
// ---------------------------------------------------------------------------
__global__ __launch_bounds__(256) void hoe_k2(
    const float* __restrict__ U, const float* __restrict__ V,
    const float* __restrict__ Dg, const float* __restrict__ Cc,
    const float* __restrict__ gamma, const float* __restrict__ beta,
    const float* __restrict__ bias_p, const float* __restrict__ W2,
    const float* __restrict__ b2, float* __restrict__ out)
{
    __shared__ float h[MM][LDH];

    const int bn = blockIdx.x >> 6;  // 0..31
    const int i  = blockIdx.x & 63;  // 0..63
    const int t  = threadIdx.x;

    // ---- phase 1: rows of h + LayerNorm (4 lanes per j, 32 f's each) ----
    const int j = t >> 2;
    const int q = t & 3;
    const bool diag = (j == i);
    const float* Urow = U  + (size_t)(bn * MM + i) * MF;
    const float* Vrow = V  + (size_t)(bn * MM + j) * MF;
    const float* Drow = Dg + (size_t)(bn * MM + i) * MF;
    const float* crow = Cc + bn * MF;

    float4 hv[8];
    float s1 = 0.f, s2 = 0.f;
#pragma unroll
    for (int u = 0; u < 8; ++u) {
        const int f = q * 32 + u * 4;
        float4 a = *(const float4*)(Urow + f);
        float4 b = *(const float4*)(Vrow + f);
        float4 c = *(const float4*)(crow + f);
        float4 v;
        v.x = a.x + b.x + c.x;  v.y = a.y + b.y + c.y;
        v.z = a.z + b.z + c.z;  v.w = a.w + b.w + c.w;
        if (diag) {
            float4 d = *(const float4*)(Drow + f);
            v.x += d.x; v.y += d.y; v.z += d.z; v.w += d.w;
        }
        hv[u] = v;
        s1 += v.x + v.y + v.z + v.w;
        s2 += v.x * v.x + v.y * v.y + v.z * v.z + v.w * v.w;
    }
    // reduce across the 4 lanes sharing one j (wave32 shuffles)
    s1 += __shfl_xor(s1, 1, 32);  s1 += __shfl_xor(s1, 2, 32);
    s2 += __shfl_xor(s2, 1, 32);  s2 += __shfl_xor(s2, 2, 32);
    const float mu   = s1 * 0.0078125f;            // /128
    const float var  = s2 * 0.0078125f - mu * mu;
    const float rstd = rsqrtf(var + 1e-5f);

#pragma unroll
    for (int u = 0; u < 8; ++u) {
        const int f = q * 32 + u * 4;
        float4 g  = *(const float4*)(gamma + f);
        float4 be = *(const float4*)(beta + f);
        float4 v  = hv[u];
        v.x = fmaxf((v.x - mu) * rstd * g.x + be.x, 0.f);
        v.y = fmaxf((v.y - mu) * rstd * g.y + be.y, 0.f);
        v.z = fmaxf((v.z - mu) * rstd * g.z + be.z, 0.f);
        v.w = fmaxf((v.w - mu) * rstd * g.w + be.w, 0.f);
        if (diag) {
            float4 bp = *(const float4*)(bias_p + f);
            v.x += bp.x; v.y += bp.y; v.z += bp.z; v.w += bp.w;
        }
        *(float4*)&h[j][f] = v;
    }
    __syncthreads();

    // ---- phase 2: out[j, d0+n] = sum_f h[j,f] * W2[d0+n, f] + b2 ----
    const int lane = t & 31;
    const int wave = t >> 5;          // column tile d0 = 16*wave
    const int n    = lane & 15;
    const int half = lane >> 4;
    const int d0   = wave * 16;

    // preload B fragments: B[k][n] = W2[d0+n, 4*ks + 2*half + {0,1}]
    v2f breg[32];
    const float* wp = W2 + (size_t)(d0 + n) * MF + 2 * half;
#pragma unroll
    for (int ks = 0; ks < 32; ++ks)
        breg[ks] = *(const v2f*)(wp + ks * 4);

    v8f acc[4] = {};
#pragma unroll
    for (int ks = 0; ks < 32; ++ks) {
#pragma unroll
        for (int jt = 0; jt < 4; ++jt) {
            v2f av = *(const v2f*)&h[jt * 16 + n][ks * 4 + 2 * half];
            acc[jt] = __builtin_amdgcn_wmma_f32_16x16x4_f32(
                        false, av, false, breg[ks], (short)0, acc[jt], false, false);
        }
    }

    const float b2v = b2[d0 + n];
    float* obase = out + (size_t)(bn * MM + i) * MM * MF + d0 + n;
#pragma unroll
    for (int jt = 0; jt < 4; ++jt) {
#pragma unroll
        for (int r = 0; r < 8; ++r) {
            int jj = jt * 16 + r + 8 * half;   // C/D layout: row r + 8*half
            obase[(size_t)jj * MF] = acc[jt][r] + b2v;
        }
    }
}

// ---------------------------------------------------------------------------
extern "C" void kernel_launch(void* const* d_in, const int* in_sizes, int n_in,
                              void* d_out, int out_size, void* d_ws, size_t ws_size,
                              hipStream_t stream)
{
    const float* x      = (const float*)d_in[0];
    const float* W1     = (const float*)d_in[1];
    const float* b1     = (const float*)d_in[2];
    const float* gamma  = (const float*)d_in[3];
    const float* beta   = (const float*)d_in[4];
    const float* bias_p = (const float*)d_in[5];
    const float* W2     = (const float*)d_in[6];
    const float* b2     = (const float*)d_in[7];
    float* out = (float*)d_out;

    // workspace layout (floats): U, V, Dg each 32*64*128; Cc 32*128  (~3.2 MB)
    float* ws = (float*)d_ws;
    float* U  = ws;
    float* V  = U  + 32 * MM * MF;
    float* Dg = V  + 32 * MM * MF;
    float* Cc = Dg + 32 * MM * MF;

    hoe_k1<<<32, 256, 0, stream>>>(x, W1, b1, U, V, Dg, Cc);
    hoe_k2<<<32 * MM, 256, 0, stream>>>(U, V, Dg, Cc, gamma, beta, bias_p, W2, b2, out);
}